// DAG_72782515798738
// MI455X (gfx1250) — compile-verified
//
#include <hip/hip_runtime.h>
#include <math.h>

// Problem constants (from reference)
#define IN_DIM  1024
#define HID     1024
#define OUT_DIM 256
#define M_DIM   1280        // HID + OUT
#define P_DIM   2304        // IN + M
#define B_DIM   8192
#define BT      16          // batch-tile rows per workgroup (WMMA M)
#define NBLK    80          // M_DIM / 16 unit-blocks
#define WAVES   8
#define TPB     256

typedef float v2f __attribute__((ext_vector_type(2)));
typedef float v8f __attribute__((ext_vector_type(8)));
typedef int   v4i __attribute__((ext_vector_type(4)));

#if defined(__has_builtin)
#if __has_builtin(__builtin_amdgcn_global_load_async_to_lds_b128)
#define HAVE_ASYNC_LDS 1
#endif
#if __has_builtin(__builtin_amdgcn_s_wait_asynccnt)
#define HAVE_WAIT_ASYNC 1
#endif
#endif

// Async global->LDS builtin takes int4 pointers: global (AS1) src, LDS (AS3) dst
typedef __attribute__((address_space(1))) v4i* g4p_t;
typedef __attribute__((address_space(3))) v4i* l4p_t;

__device__ __forceinline__ v8f wmma_f32(v2f a, v2f b, v8f c) {
    // (neg_a, A, neg_b, B, c_mod, C, reuse_a, reuse_b)
    return __builtin_amdgcn_wmma_f32_16x16x4_f32(false, a, false, b,
                                                 (short)0, c, false, false);
}

__global__ __launch_bounds__(TPB)
void dag_recurrence_kernel(const float* __restrict__ x,
                           const float* __restrict__ W,
                           const float* __restrict__ bias,
                           float* __restrict__ out) {
    // x-tile staged once via async global->LDS (64 KB) + 16x1280 f32
    // state/accumulator tile (80 KB): 144 KB of the 320 KB WGP LDS pool.
    __shared__ float sX[BT * IN_DIM];
    __shared__ float sA[BT * M_DIM];

    const int tid    = threadIdx.x;
    const int wv     = tid >> 5;          // wave id 0..7
    const int lane   = tid & 31;
    const int row16  = lane & 15;         // WMMA row / col index within tile
    const int khalf  = lane >> 4;         // 0: K pair {0,1}, 1: K pair {2,3}
    const int rowbase = blockIdx.x * BT;  // first batch row of this workgroup

    // ---- Stage x tile: one contiguous 64 KB block (16 consecutive rows) ----
    {
        const float* gsrc = x + (size_t)rowbase * IN_DIM;
#ifdef HAVE_ASYNC_LDS
        #pragma unroll
        for (int i = 0; i < 16; ++i) {
            const int idx = (tid + i * TPB) * 4;   // float index, 16B granules
            __builtin_amdgcn_global_load_async_to_lds_b128(
                (g4p_t)(gsrc + idx), (l4p_t)&sX[idx], 0, 0);
        }
#ifdef HAVE_WAIT_ASYNC
        __builtin_amdgcn_s_wait_asynccnt(0);
#endif
#else
        #pragma unroll
        for (int i = 0; i < 16; ++i) {
            const int idx = (tid + i * TPB) * 4;
            *(float4*)&sX[idx] = *(const float4*)(gsrc + idx);
        }
#endif
    }
    __syncthreads();

    // ---------------- Phase A: sA = x_tile @ Wx^T + bias ----------------
    // Each wave owns 10 column tiles (160 units). K = 1024 in chunks of 4.
    for (int nt = 0; nt < 10; ++nt) {
        const int col0 = (wv * 10 + nt) * 16;
        v8f c = {};
        const float* xls  = &sX[row16 * IN_DIM + 2 * khalf];           // LDS
        const float* wrow = W + (size_t)(col0 + row16) * P_DIM + 2 * khalf;
        #pragma unroll 4
        for (int k0 = 0; k0 < IN_DIM; k0 += 4) {
            v2f a  = *(const v2f*)(xls + k0);    // A[m][k] from LDS
            v2f bb = *(const v2f*)(wrow + k0);   // B[k][n] from L2-resident W
            c = wmma_f32(a, bb, c);
        }
        const float bv = bias[col0 + row16];
        #pragma unroll
        for (int v = 0; v < 8; ++v) {
            // C layout: VGPR v -> row v + 8*khalf, col = lane%16
            sA[(v + 8 * khalf) * M_DIM + col0 + row16] = c[v] + bv;
        }
    }
    __syncthreads();

    // ------- Phase B: blocked sequential recurrence over 80 blocks -------
    for (int t = 0; t < NBLK; ++t) {
        const int base = t * 16;

        // Intra-block sequential triangular solve: one wave, lanes 0..15
        if (wv == (t & 7) && lane < 16) {
            float a[16];
            #pragma unroll
            for (int u = 0; u < 16; ++u)
                a[u] = sA[lane * M_DIM + base + u];
            #pragma unroll
            for (int jj = 0; jj < 16; ++jj) {
                float s = tanhf(a[jj]);
                a[jj] = s;
                #pragma unroll
                for (int kk = jj + 1; kk < 16; ++kk) {
                    // uniform address -> scalar load from L2-resident W
                    a[kk] += s * W[(size_t)(base + kk) * P_DIM + IN_DIM + base + jj];
                }
            }
            #pragma unroll
            for (int u = 0; u < 16; ++u)
                sA[lane * M_DIM + base + u] = a[u];
        }
        __syncthreads();

        // s_t A-operand chunks are invariant across the trailing loop: hoist.
        v2f a0[4];
        #pragma unroll
        for (int ch = 0; ch < 4; ++ch)
            a0[ch] = *(const v2f*)&sA[row16 * M_DIM + base + ch * 4 + 2 * khalf];

        // Trailing update: A[:, block f] += s_t @ L_sub^T, f > t,
        // round-robin over the 8 waves (wave-uniform loop -> EXEC all ones)
        for (int f = t + 1 + wv; f < NBLK; f += WAVES) {
            v8f c;
            #pragma unroll
            for (int v = 0; v < 8; ++v)
                c[v] = sA[(v + 8 * khalf) * M_DIM + f * 16 + row16];
            const float* wrow = W + (size_t)(f * 16 + row16) * P_DIM
                                  + IN_DIM + base + 2 * khalf;
            #pragma unroll
            for (int ch = 0; ch < 4; ++ch) {
                v2f bb = *(const v2f*)(wrow + ch * 4);
                c = wmma_f32(a0[ch], bb, c);
            }
            #pragma unroll
            for (int v = 0; v < 8; ++v)
                sA[(v + 8 * khalf) * M_DIM + f * 16 + row16] = c[v];
        }
        __syncthreads();
    }

    // -------- Epilogue: out = sigmoid(states[:, IN+HID:]) --------
    for (int i = tid; i < BT * OUT_DIM; i += TPB) {
        const int r = i >> 8;       // OUT_DIM == 256
        const int o = i & 255;
        float v = sA[r * M_DIM + HID + o];
        out[(size_t)(rowbase + r) * OUT_DIM + o] = 1.0f / (1.0f + __expf(-v));
    }
}

extern "C" void kernel_launch(void* const* d_in, const int* in_sizes, int n_in,
                              void* d_out, int out_size, void* d_ws, size_t ws_size,
                              hipStream_t stream) {
    const float* x = (const float*)d_in[0];   // (8192, 1024) f32
    const float* W = (const float*)d_in[1];   // (1280, 2304) f32
    const float* b = (const float*)d_in[2];   // (1280,) f32
    float* out = (float*)d_out;               // (8192, 256) f32

    dim3 grid(B_DIM / BT);   // 512 workgroups, one 16-row batch tile each
    dim3 block(TPB);         // 8 waves of 32
    dag_recurrence_kernel<<<grid, block, 0, stream>>>(x, W, b, out);
}